// RejectionSampler_12043088298606
// MI455X (gfx1250) — compile-verified
//
#include <hip/hip_runtime.h>
#include <hip/hip_bf16.h>
#include <math.h>

// ---------------------------------------------------------------------------
// Rejection sampler (speculative decoding) for MI455X / gfx1250.
//
// Memory-bound streamer: ~142 MB of f32 reads -> ~6.1 us at 23.3 TB/s.
// Per logits row (one workgroup = 8 wave32s) we compute in ONE pass:
//   m  = max_v(x_v / T)
//   s  = sum_v exp(x_v/T - m)            (online softmax)
//   argmax_{v != draft} (x_v/T - log q_v)   == argmax(masked_prob / q)
//   t_d = x_draft / T                     -> p_draft = exp(t_d - m)/s
// The argmax identity removes the need to materialize softmax for sampling.
//
// CDNA5 path: per-wave, 3-deep rotating GLOBAL_LOAD_ASYNC_TO_LDS_B128 pipeline
// (ASYNCcnt-tracked) keeps two tiles of prefetch distance ahead of the
// v_exp/v_log consumer, decoupling HBM latency without VGPR cost.
// ---------------------------------------------------------------------------

#define BLOCK 256
#define WAVES 8                // 256 threads / wave32
#define WTILE 128              // floats per wave-tile (32 lanes * 4)
#define TILE  (WAVES * WTILE)  // 1024 floats per block-tile
#define PIPE  3                // tiles in flight (rotating LDS buffers)

#if defined(__HIP_DEVICE_COMPILE__) && __has_builtin(__builtin_amdgcn_global_load_async_to_lds_b128)
#define USE_ASYNC 1
#else
#define USE_ASYNC 0
#endif

typedef int v4i __attribute__((vector_size(16)));
typedef __attribute__((address_space(1))) v4i g_v4i;   // global
typedef __attribute__((address_space(3))) v4i l_v4i;   // LDS

__device__ __forceinline__ void cfence() { asm volatile("" ::: "memory"); }

__device__ __forceinline__ void async_copy16(const float* gsrc, float* ldst) {
#if USE_ASYNC
  __builtin_amdgcn_global_load_async_to_lds_b128(
      (g_v4i*)gsrc, (l_v4i*)ldst, /*imm offset*/ 0, /*cpol*/ 0);
#else
  (void)gsrc; (void)ldst;
#endif
}

// ASYNCcnt completes in order; wait until at most `N` ops remain outstanding.
// Builtin needs literal immediates; compiler fence stops LDS reads from being
// scheduled above the completion wait.
#if USE_ASYNC
__device__ __forceinline__ void wait_async_4() { __builtin_amdgcn_s_wait_asynccnt(4); cfence(); }
__device__ __forceinline__ void wait_async_2() { __builtin_amdgcn_s_wait_asynccnt(2); cfence(); }
__device__ __forceinline__ void wait_async_0() { __builtin_amdgcn_s_wait_asynccnt(0); cfence(); }
#endif

struct RowState {
  float m, s;      // online softmax: m = running max of x/T, s = sum exp(x/T - m)
  float bv;        // best score  (x/T - log q), draft id masked out
  int   bi;        // best index
  float td;        // x/T at the draft token id
  int   saw;       // lane saw the draft id
};

__device__ __forceinline__ void upd(RowState& st, float x, float qv, int v,
                                    float invT, int draft) {
  float t = x * invT;
  // online softmax update (rescale on new max)
  if (t > st.m) { st.s = st.s * __expf(st.m - t) + 1.0f; st.m = t; }
  else          { st.s += __expf(t - st.m); }
  if (v != draft) {
    float sc = t - __logf(qv);
    if (sc > st.bv || (sc == st.bv && v < st.bi)) { st.bv = sc; st.bi = v; }
  } else {
    st.td = t; st.saw = 1;
  }
}

__device__ __forceinline__ void load4_unaligned(const float* p, float out[4]) {
  // rows are V=50257 floats long -> row bases are only 4B aligned.
  __builtin_memcpy(out, p, 16);
}

__global__ void __launch_bounds__(BLOCK)
rs_row_kernel(const float* __restrict__ logits,
              const float* __restrict__ temps,
              const int*   __restrict__ draft_ids,
              const float* __restrict__ u,
              const float* __restrict__ q,
              const float* __restrict__ q_bonus,
              int* __restrict__ accept_ws,   // [T]
              int* __restrict__ rec_ws,      // [T]
              int* __restrict__ bonus_ws,    // [B]
              int B, int K, int V)
{
  const int row  = blockIdx.x;
  const int T    = B * K;
  const int tid  = threadIdx.x;
  const int w    = tid >> 5;
  const int lane = tid & 31;

  const bool isDraft = (row < T);
  const int  b       = isDraft ? (row / K) : (row - T);
  const int  draft   = isDraft ? draft_ids[row] : -1;
  const float invT   = 1.0f / temps[b];

  const float* __restrict__ rowL = logits + (size_t)row * (size_t)V;
  const float* __restrict__ rowQ = (isDraft ? q : q_bonus) + (size_t)b * (size_t)V;

  __shared__ __align__(16) float sL[PIPE][WAVES][WTILE];
  __shared__ __align__(16) float sQ[PIPE][WAVES][WTILE];
  __shared__ float redM[WAVES], redS[WAVES], redV[WAVES];
  __shared__ int   redI[WAVES];
  __shared__ float sTd;

  RowState st;
  st.m = -INFINITY; st.s = 0.0f; st.bv = -INFINITY; st.bi = 0; st.td = 0.0f; st.saw = 0;

  const int nFull    = V / TILE;           // 49 full tiles for V=50257
  const int laneBase = w * WTILE + lane * 4;

#if USE_ASYNC
  // --- 3-deep rotating async global->LDS pipeline, per-wave self-consumed ---
  {
    const int npre = (nFull < (PIPE - 1)) ? nFull : (PIPE - 1);
    for (int p = 0; p < npre; ++p) {
      const int off = p * TILE + laneBase;
      async_copy16(rowL + off, &sL[p][w][lane * 4]);
      async_copy16(rowQ + off, &sQ[p][w][lane * 4]);
    }
  }
  int buf  = 0;                       // buffer holding tile t
  int abuf = (PIPE - 1) % PIPE;       // buffer for tile t+PIPE-1
  for (int t = 0; t < nFull; ++t) {
    const int ahead = t + (PIPE - 1);
    if (ahead < nFull) {
      const int off = ahead * TILE + laneBase;
      async_copy16(rowL + off, &sL[abuf][w][lane * 4]);
      async_copy16(rowQ + off, &sQ[abuf][w][lane * 4]);
    }
    // #tiles issued after tile t (2 async ops each) -> exact wait threshold
    const int rem  = nFull - 1 - t;
    const int pend = (rem < (PIPE - 1)) ? rem : (PIPE - 1);
    if      (pend >= 2) wait_async_4();
    else if (pend == 1) wait_async_2();
    else                wait_async_0();

    const float4 lx = *(const float4*)&sL[buf][w][lane * 4];
    const float4 lq = *(const float4*)&sQ[buf][w][lane * 4];
    const int v0 = t * TILE + laneBase;
    upd(st, lx.x, lq.x, v0 + 0, invT, draft);
    upd(st, lx.y, lq.y, v0 + 1, invT, draft);
    upd(st, lx.z, lq.z, v0 + 2, invT, draft);
    upd(st, lx.w, lq.w, v0 + 3, invT, draft);

    buf  = (buf  + 1 == PIPE) ? 0 : buf  + 1;
    abuf = (abuf + 1 == PIPE) ? 0 : abuf + 1;
  }
#else
  for (int t = 0; t < nFull; ++t) {
    const int v0 = t * TILE + laneBase;
    float lx[4], lq[4];
    load4_unaligned(rowL + v0, lx);
    load4_unaligned(rowQ + v0, lq);
    upd(st, lx[0], lq[0], v0 + 0, invT, draft);
    upd(st, lx[1], lq[1], v0 + 1, invT, draft);
    upd(st, lx[2], lq[2], v0 + 2, invT, draft);
    upd(st, lx[3], lq[3], v0 + 3, invT, draft);
  }
#endif

  // tail (V % 1024 = 81 elements), scalar guarded
  for (int v = nFull * TILE + tid; v < V; v += BLOCK) {
    upd(st, rowL[v], rowQ[v], v, invT, draft);
  }

  // --- wave32 reduction (shfl_xor tree) ---
  for (int off = 16; off > 0; off >>= 1) {
    const float om = __shfl_xor(st.m,  off, 32);
    const float os = __shfl_xor(st.s,  off, 32);
    const float ov = __shfl_xor(st.bv, off, 32);
    const int   oi = __shfl_xor(st.bi, off, 32);
    const float M = fmaxf(st.m, om);
    st.s = st.s * __expf(st.m - M) + os * __expf(om - M);
    st.m = M;
    if (ov > st.bv || (ov == st.bv && oi < st.bi)) { st.bv = ov; st.bi = oi; }
  }
  if (lane == 0) { redM[w] = st.m; redS[w] = st.s; redV[w] = st.bv; redI[w] = st.bi; }
  if (st.saw) sTd = st.td;   // exactly one lane per draft row hits the draft id
  __syncthreads();

  if (tid == 0) {
    float m = redM[0], s = redS[0], bv = redV[0];
    int bi = redI[0];
    for (int i = 1; i < WAVES; ++i) {
      const float M = fmaxf(m, redM[i]);
      s = s * __expf(m - M) + redS[i] * __expf(redM[i] - M);
      m = M;
      if (redV[i] > bv || (redV[i] == bv && redI[i] < bi)) { bv = redV[i]; bi = redI[i]; }
    }
    if (isDraft) {
      const float p = __expf(sTd - m) / s;   // target prob at draft id
      accept_ws[row] = (p >= u[row]) ? 1 : 0;
      rec_ws[row]    = bi;
    } else {
      bonus_ws[b] = bi;
    }
  }
}

__global__ void rs_finalize_kernel(const int* __restrict__ draft_ids,
                                   const int* __restrict__ accept_ws,
                                   const int* __restrict__ rec_ws,
                                   const int* __restrict__ bonus_ws,
                                   int* __restrict__ out,
                                   int B, int K)
{
  const int b = blockIdx.x * blockDim.x + threadIdx.x;
  if (b >= B) return;
  int r = K;
  for (int k = 0; k < K; ++k) {
    if (!accept_ws[b * K + k]) { r = k; break; }
  }
  for (int k = 0; k < K; ++k) {
    int tok;
    if      (k < r)  tok = draft_ids[b * K + k];
    else if (k == r) tok = rec_ws[b * K + k];
    else             tok = -1;
    out[b * (K + 1) + k] = tok;
  }
  out[b * (K + 1) + K] = (r == K) ? bonus_ws[b] : -1;
}

extern "C" void kernel_launch(void* const* d_in, const int* in_sizes, int n_in,
                              void* d_out, int out_size, void* d_ws, size_t ws_size,
                              hipStream_t stream) {
  const int*   draft  = (const int*)d_in[0];     // [B,K]
  const float* logits = (const float*)d_in[1];   // [T+B, V]
  const float* temps  = (const float*)d_in[2];   // [B]
  const float* u      = (const float*)d_in[3];   // [T]
  const float* q      = (const float*)d_in[4];   // [B,V]
  const float* qb     = (const float*)d_in[5];   // [B,V]

  const int B = in_sizes[2];
  const int K = in_sizes[0] / B;
  const int T = B * K;
  const int V = in_sizes[1] / (T + B);

  int* accept_ws = (int*)d_ws;
  int* rec_ws    = accept_ws + T;
  int* bonus_ws  = rec_ws + T;
  int* out       = (int*)d_out;   // reference output dtype is int32

  hipLaunchKernelGGL(rs_row_kernel, dim3(T + B), dim3(BLOCK), 0, stream,
                     logits, temps, draft, u, q, qb,
                     accept_ws, rec_ws, bonus_ws, B, K, V);
  hipLaunchKernelGGL(rs_finalize_kernel, dim3((B + 63) / 64), dim3(64), 0, stream,
                     draft, accept_ws, rec_ws, bonus_ws, out, B, K);
}